// SelectiveCopyingS5_29429115912325
// MI455X (gfx1250) — compile-verified
//
#include <hip/hip_runtime.h>
#include <hip/hip_bf16.h>
#include <math.h>

// Problem constants (from reference)
#define B_SZ   32
#define LEN    4096
#define HDIM   512
#define PDIM   384
#define VOC    32
#define OUTL   16
#define NCHUNK 16
#define LCHUNK 256   // LEN / NCHUNK

typedef __attribute__((ext_vector_type(2))) float v2f;
typedef __attribute__((ext_vector_type(8))) float v8f;

static __device__ __forceinline__ v8f wmma4(v2f a, v2f b, v8f c) {
    return __builtin_amdgcn_wmma_f32_16x16x4_f32(false, a, false, b,
                                                 (short)0, c, false, false);
}

// ---------------------------------------------------------------------------
// K1: precompute Lbar[p] and the (V x P) complex table
//     tab[v,p] = ((Lbar[p]-1)/Lambda[p]) * sum_h embed[v,h]*(B_re+i*B_im)[p,h]
// ---------------------------------------------------------------------------
__global__ void k_precompute(const float* __restrict__ embed,
                             const float* __restrict__ Lre, const float* __restrict__ Lim,
                             const float* __restrict__ Bre, const float* __restrict__ Bim,
                             const float* __restrict__ lstep,
                             float* __restrict__ lbr, float* __restrict__ lbi,
                             float* __restrict__ tre, float* __restrict__ tim) {
    int idx = blockIdx.x * blockDim.x + threadIdx.x;
    if (idx >= VOC * PDIM) return;
    int p = idx % PDIM;
    int v = idx / PDIM;

    float lre = Lre[p], lim = Lim[p];
    float st  = expf(lstep[p]);
    float er  = expf(lre * st);
    float sw, cw;
    sincosf(lim * st, &sw, &cw);
    float lbar_r = er * cw;
    float lbar_i = er * sw;
    if (v == 0) { lbr[p] = lbar_r; lbi[p] = lbar_i; }

    // scale = (Lbar - 1) / Lambda   (complex division; lre <= -0.001 so den>0)
    float nr  = lbar_r - 1.0f, ni = lbar_i;
    float den = lre * lre + lim * lim;
    float sr  = (nr * lre + ni * lim) / den;
    float si  = (ni * lre - nr * lim) / den;

    float ar = 0.0f, ai = 0.0f;
    const float* eb = embed + v * HDIM;
    const float* br = Bre + p * HDIM;
    const float* bi = Bim + p * HDIM;
#pragma unroll 4
    for (int h = 0; h < HDIM; ++h) {
        float u = eb[h];
        ar = fmaf(u, br[h], ar);
        ai = fmaf(u, bi[h], ai);
    }
    tre[v * PDIM + p] = sr * ar - si * ai;
    tim[v * PDIM + p] = sr * ai + si * ar;
}

// ---------------------------------------------------------------------------
// K1b: CT_re[k,h] =  2*C_re[h,k],  CT_im[k,h] = -2*C_im[h,k]
//      (folds the 2*Re(.) factor and the minus sign; makes GEMM-B coalesced)
// ---------------------------------------------------------------------------
__global__ void k_transposeC(const float* __restrict__ Cre, const float* __restrict__ Cim,
                             float* __restrict__ CreT, float* __restrict__ CimT) {
    int idx = blockIdx.x * blockDim.x + threadIdx.x;
    if (idx >= HDIM * PDIM) return;
    int h = idx % HDIM;
    int k = idx / HDIM;
    CreT[idx] =  2.0f * Cre[h * PDIM + k];
    CimT[idx] = -2.0f * Cim[h * PDIM + k];
}

// ---------------------------------------------------------------------------
// K2: per-chunk partial scans (chunks 0..14), zero initial state per chunk.
// ---------------------------------------------------------------------------
__global__ void k_chunks(const int* __restrict__ tokens,
                         const float* __restrict__ lbr, const float* __restrict__ lbi,
                         const float* __restrict__ tre, const float* __restrict__ tim,
                         float* __restrict__ pre, float* __restrict__ pim) {
    int idx = blockIdx.x * blockDim.x + threadIdx.x;
    if (idx >= (NCHUNK - 1) * B_SZ * PDIM) return;
    int p = idx % PDIM;
    int b = (idx / PDIM) % B_SZ;
    int c = idx / (PDIM * B_SZ);

    float lr = lbr[p], li = lbi[p];
    float xr = 0.0f, xi = 0.0f;
    const int* tk = tokens + b * LEN + c * LCHUNK;
#pragma unroll 4
    for (int l = 0; l < LCHUNK; ++l) {
        int tok = tk[l];                          // uniform across wave -> broadcast
        float brv = tre[tok * PDIM + p];          // coalesced in p
        float biv = tim[tok * PDIM + p];
        float nr = fmaf(lr, xr, fmaf(-li, xi, brv));
        float ni = fmaf(lr, xi, fmaf( li, xr, biv));
        xr = nr; xi = ni;
    }
    pre[idx] = xr;
    pim[idx] = xi;
}

// ---------------------------------------------------------------------------
// K3: combine chunk partials (W = Lbar^256 by repeated squaring), then run the
//     last 256 steps sequentially, storing xs for the final 16 positions.
// ---------------------------------------------------------------------------
__global__ void k_combine_tail(const int* __restrict__ tokens,
                               const float* __restrict__ lbr, const float* __restrict__ lbi,
                               const float* __restrict__ tre, const float* __restrict__ tim,
                               const float* __restrict__ pre, const float* __restrict__ pim,
                               float* __restrict__ xre, float* __restrict__ xim) {
    int idx = blockIdx.x * blockDim.x + threadIdx.x;
    if (idx >= B_SZ * PDIM) return;
    int p = idx % PDIM;
    int b = idx / PDIM;

    float lr = lbr[p], li = lbi[p];
    // W = Lbar^LCHUNK via 8 complex squarings (LCHUNK = 2^8)
    float wr = lr, wi = li;
#pragma unroll
    for (int s = 0; s < 8; ++s) {
        float nr = wr * wr - wi * wi;
        float ni = 2.0f * wr * wi;
        wr = nr; wi = ni;
    }
    // combine chunks 0..14 -> state after l = 15*256 - 1 = 3839
    float xr = 0.0f, xi = 0.0f;
#pragma unroll
    for (int c = 0; c < NCHUNK - 1; ++c) {
        float prv = pre[(c * B_SZ + b) * PDIM + p];
        float piv = pim[(c * B_SZ + b) * PDIM + p];
        float nr = fmaf(wr, xr, fmaf(-wi, xi, prv));
        float ni = fmaf(wr, xi, fmaf( wi, xr, piv));
        xr = nr; xi = ni;
    }
    // sequential tail over the last chunk; store the final 16 states
    const int* tk = tokens + b * LEN + (NCHUNK - 1) * LCHUNK;
#pragma unroll 4
    for (int l = 0; l < LCHUNK; ++l) {
        int tok = tk[l];
        float brv = tre[tok * PDIM + p];
        float biv = tim[tok * PDIM + p];
        float nr = fmaf(lr, xr, fmaf(-li, xi, brv));
        float ni = fmaf(lr, xi, fmaf( li, xr, biv));
        xr = nr; xi = ni;
        if (l >= LCHUNK - OUTL) {
            int t = l - (LCHUNK - OUTL);
            xre[(b * OUTL + t) * PDIM + p] = xr;
            xim[(b * OUTL + t) * PDIM + p] = xi;
        }
    }
}

// ---------------------------------------------------------------------------
// K4: y[r,h] = (Xre @ 2CreT + Xim @ (-2CimT))[r,h] + embed[tok(r), h] * D[h]
//     M=512, N=512, two K=384 streams; one 16x16 tile per wave,
//     V_WMMA_F32_16X16X4_F32, dual accumulators.
// ---------------------------------------------------------------------------
__global__ void k_ygemm(const float* __restrict__ xre, const float* __restrict__ xim,
                        const float* __restrict__ CreT, const float* __restrict__ CimT,
                        const int* __restrict__ tokens, const float* __restrict__ embed,
                        const float* __restrict__ Dv, float* __restrict__ y) {
    const int lane = threadIdx.x & 31;
    const int wave = threadIdx.x >> 5;
    const int tile = blockIdx.x * 4 + wave;   // 1024 tiles: 32 (M) x 32 (N)
    const int tn = tile & 31;
    const int tm = tile >> 5;
    const int rA   = tm * 16 + (lane & 15);   // A: both lane halves hold M=0..15
    const int nB   = tn * 16 + (lane & 15);   // B: column per lane (coalesced)
    const int koff = (lane >> 4) * 2;         // low lanes {k,k+1}, high {k+2,k+3}

    const float* Apr = xre  + rA * PDIM + koff;
    const float* Api = xim  + rA * PDIM + koff;
    const float* Bpr = CreT + koff * HDIM + nB;
    const float* Bpi = CimT + koff * HDIM + nB;

    v8f acc0 = {}, acc1 = {};
#pragma unroll 4
    for (int k0 = 0; k0 < PDIM; k0 += 4) {
        v2f ar = *(const v2f*)(Apr + k0);
        v2f br; br.x = Bpr[k0 * HDIM]; br.y = Bpr[(k0 + 1) * HDIM];
        acc0 = wmma4(ar, br, acc0);
        v2f ai = *(const v2f*)(Api + k0);
        v2f bi; bi.x = Bpi[k0 * HDIM]; bi.y = Bpi[(k0 + 1) * HDIM];
        acc1 = wmma4(ai, bi, acc1);
    }
    v8f acc = acc0 + acc1;

    const int col = tn * 16 + (lane & 15);
    const float dcol = Dv[col];
#pragma unroll
    for (int v = 0; v < 8; ++v) {
        int row = tm * 16 + v + ((lane >> 4) << 3);   // C/D: VGPR v -> rows v, v+8
        int bb = row >> 4, tt = row & 15;
        int tok = tokens[bb * LEN + (LEN - OUTL + tt)];
        y[row * HDIM + col] = acc[v] + embed[tok * HDIM + col] * dcol;
    }
}

// ---------------------------------------------------------------------------
// K5: h1 = relu(y @ W1 + b1)    M=512, N=512, K=512
// ---------------------------------------------------------------------------
__global__ void k_mlp1(const float* __restrict__ y, const float* __restrict__ W1,
                       const float* __restrict__ b1, float* __restrict__ h1) {
    const int lane = threadIdx.x & 31;
    const int wave = threadIdx.x >> 5;
    const int tile = blockIdx.x * 4 + wave;   // 1024 tiles
    const int tn = tile & 31;
    const int tm = tile >> 5;
    const int rA   = tm * 16 + (lane & 15);
    const int nB   = tn * 16 + (lane & 15);
    const int koff = (lane >> 4) * 2;

    const float* Ap = y  + rA * HDIM + koff;
    const float* Bp = W1 + koff * HDIM + nB;

    v8f acc0 = {}, acc1 = {};
#pragma unroll 4
    for (int k0 = 0; k0 < HDIM; k0 += 8) {
        v2f a0 = *(const v2f*)(Ap + k0);
        v2f b0; b0.x = Bp[k0 * HDIM]; b0.y = Bp[(k0 + 1) * HDIM];
        acc0 = wmma4(a0, b0, acc0);
        v2f a1 = *(const v2f*)(Ap + k0 + 4);
        v2f b1v; b1v.x = Bp[(k0 + 4) * HDIM]; b1v.y = Bp[(k0 + 5) * HDIM];
        acc1 = wmma4(a1, b1v, acc1);
    }
    v8f acc = acc0 + acc1;

    const int col = tn * 16 + (lane & 15);
    const float bias = b1[col];
#pragma unroll
    for (int v = 0; v < 8; ++v) {
        int row = tm * 16 + v + ((lane >> 4) << 3);
        float r = acc[v] + bias;
        h1[row * HDIM + col] = r > 0.0f ? r : 0.0f;
    }
}

// ---------------------------------------------------------------------------
// K6: out = h1 @ W2 + b2    M=512, N=32, K=512  -> d_out (B, 16, V) f32
// ---------------------------------------------------------------------------
__global__ void k_mlp2(const float* __restrict__ h1, const float* __restrict__ W2,
                       const float* __restrict__ b2, float* __restrict__ out) {
    const int lane = threadIdx.x & 31;
    const int wave = threadIdx.x >> 5;
    const int tile = blockIdx.x * 4 + wave;   // 64 tiles: 32 (M) x 2 (N)
    const int tn = tile & 1;
    const int tm = tile >> 1;
    const int rA   = tm * 16 + (lane & 15);
    const int nB   = tn * 16 + (lane & 15);
    const int koff = (lane >> 4) * 2;

    const float* Ap = h1 + rA * HDIM + koff;
    const float* Bp = W2 + koff * VOC + nB;

    v8f acc0 = {}, acc1 = {};
#pragma unroll 4
    for (int k0 = 0; k0 < HDIM; k0 += 8) {
        v2f a0 = *(const v2f*)(Ap + k0);
        v2f b0; b0.x = Bp[k0 * VOC]; b0.y = Bp[(k0 + 1) * VOC];
        acc0 = wmma4(a0, b0, acc0);
        v2f a1 = *(const v2f*)(Ap + k0 + 4);
        v2f b1v; b1v.x = Bp[(k0 + 4) * VOC]; b1v.y = Bp[(k0 + 5) * VOC];
        acc1 = wmma4(a1, b1v, acc1);
    }
    v8f acc = acc0 + acc1;

    const int col = tn * 16 + (lane & 15);
    const float bias = b2[col];
#pragma unroll
    for (int v = 0; v < 8; ++v) {
        int row = tm * 16 + v + ((lane >> 4) << 3);
        out[row * VOC + col] = acc[v] + bias;
    }
}

// ---------------------------------------------------------------------------
extern "C" void kernel_launch(void* const* d_in, const int* in_sizes, int n_in,
                              void* d_out, int out_size, void* d_ws, size_t ws_size,
                              hipStream_t stream) {
    const int*   tokens = (const int*)  d_in[0];
    const float* embed  = (const float*)d_in[1];
    const float* Lre    = (const float*)d_in[2];
    const float* Lim    = (const float*)d_in[3];
    const float* Bre    = (const float*)d_in[4];
    const float* Bim    = (const float*)d_in[5];
    const float* Cre    = (const float*)d_in[6];
    const float* Cim    = (const float*)d_in[7];
    const float* Dv     = (const float*)d_in[8];
    const float* lstep  = (const float*)d_in[9];
    const float* W1     = (const float*)d_in[10];
    const float* b1     = (const float*)d_in[11];
    const float* W2     = (const float*)d_in[12];
    const float* b2     = (const float*)d_in[13];
    float* out = (float*)d_out;

    // Workspace carve (floats). Total ~6.9 MB. All offsets even -> 8B aligned.
    float* w    = (float*)d_ws;
    float* lbr  = w; w += PDIM;
    float* lbi  = w; w += PDIM;
    float* tre  = w; w += VOC * PDIM;
    float* tim  = w; w += VOC * PDIM;
    float* pre  = w; w += (NCHUNK - 1) * B_SZ * PDIM;
    float* pim  = w; w += (NCHUNK - 1) * B_SZ * PDIM;
    float* xre  = w; w += B_SZ * OUTL * PDIM;
    float* xim  = w; w += B_SZ * OUTL * PDIM;
    float* CreT = w; w += PDIM * HDIM;
    float* CimT = w; w += PDIM * HDIM;
    float* yb   = w; w += B_SZ * OUTL * HDIM;
    float* h1   = w; w += B_SZ * OUTL * HDIM;

    k_precompute<<<(VOC * PDIM + 255) / 256, 256, 0, stream>>>(
        embed, Lre, Lim, Bre, Bim, lstep, lbr, lbi, tre, tim);

    k_transposeC<<<(HDIM * PDIM + 255) / 256, 256, 0, stream>>>(
        Cre, Cim, CreT, CimT);

    k_chunks<<<((NCHUNK - 1) * B_SZ * PDIM + 255) / 256, 256, 0, stream>>>(
        tokens, lbr, lbi, tre, tim, pre, pim);

    k_combine_tail<<<(B_SZ * PDIM + 255) / 256, 256, 0, stream>>>(
        tokens, lbr, lbi, tre, tim, pre, pim, xre, xim);

    k_ygemm<<<256, 128, 0, stream>>>(xre, xim, CreT, CimT, tokens, embed, Dv, yb);

    k_mlp1<<<256, 128, 0, stream>>>(yb, W1, b1, h1);

    k_mlp2<<<16, 128, 0, stream>>>(h1, W2, b2, out);
}